// GraphAttention_7859790151959
// MI455X (gfx1250) — compile-verified
//
#include <hip/hip_runtime.h>
#include <hip/hip_bf16.h>

typedef float v2f __attribute__((ext_vector_type(2)));
typedef float v8f __attribute__((ext_vector_type(8)));

#define BSZ 2
#define NN 512
#define NNP 514        // padded LDS row stride (dwords): 514 % 64 == 2 -> bank spread, 8B aligned
#define DIM 512
#define NH 16
#define DD 32
#define EPSV 1e-5f
#define XPAD 516       // kernel-1 X-tile stride: %64==4 bank spread, 16B aligned

// ---------------------------------------------------------------------------
// Kernel 1: fused 3x projection GEMM  Y = X @ W^T + b  (fp32 WMMA 16x16x4)
// Block = 8 waves sharing one M-tile: X tile (16x512) staged once into LDS,
// each wave owns a distinct 16-wide N-tile (A from LDS, B rows of W from
// global).  A layout (32b 16x4): lane l(0..15): M=l, v0=K0,v1=K1; lanes
// 16..31 hold K2/K3.  C/D: vgpr r, lane l -> M = r + 8*(l>>4), N = l&15.
// ---------------------------------------------------------------------------
__global__ __launch_bounds__(256) void proj_gemm_kernel(
    const float* __restrict__ X,
    const float* __restrict__ Wfc,  const float* __restrict__ bfc,
    const float* __restrict__ Wfc2, const float* __restrict__ bfc2,
    const float* __restrict__ Wval, const float* __restrict__ bval,
    float* __restrict__ Pfc, float* __restrict__ Pfc2, float* __restrict__ Pval)
{
    __shared__ float s_x[16 * XPAD];            // 33 KB X tile (padded rows)

    const int tid  = threadIdx.x;
    const int lane = tid & 31;
    const int wave = tid >> 5;

    const int blk = blockIdx.x;                 // 768 blocks = 3 mats * 256
    const int mat = blk >> 8;
    const int rem = blk & 255;                  // 64 mtiles x 4 ot-groups
    const int mt  = rem >> 2;
    const int ot  = (rem & 3) * 8 + wave;       // this wave's N tile (0..31)

    const float* Wm = (mat == 0) ? Wfc : (mat == 1) ? Wfc2 : Wval;
    const float* bm = (mat == 0) ? bfc : (mat == 1) ? bfc2 : bval;
    float*       Ym = (mat == 0) ? Pfc : (mat == 1) ? Pfc2 : Pval;

    // ---- cooperative stage of the shared X tile [16][512] ------------------
    const float* xbase = X + (size_t)(mt * 16) * DIM;
    for (int idx = tid; idx < 16 * 128; idx += 256) {      // 2048 float4
        const int i  = idx >> 7;
        const int kq = idx & 127;
        *(float4*)(s_x + i * XPAD + kq * 4) =
            *(const float4*)(xbase + (size_t)i * DIM + kq * 4);
    }
    __syncthreads();

    const int l16 = lane & 15;
    const int hh  = lane >> 4;                  // K-pair select (0 or 1)

    const float* arow = s_x + l16 * XPAD + 2 * hh;            // A from LDS
    const float* brow = Wm + (size_t)(ot * 16 + l16) * DIM;   // B[k,N]=W[N,k]
    __builtin_prefetch(brow, 0, 1);                           // global_prefetch_b8

    v8f c = {0.f, 0.f, 0.f, 0.f, 0.f, 0.f, 0.f, 0.f};
    #pragma unroll 8
    for (int k0 = 0; k0 < DIM; k0 += 4) {
        v2f a = *(const v2f*)(arow + k0);
        v2f b = *(const v2f*)(brow + k0 + 2 * hh);
        c = __builtin_amdgcn_wmma_f32_16x16x4_f32(false, a, false, b,
                                                  (short)0, c, false, false);
    }

    const int   ocol = ot * 16 + l16;
    const float bias = bm[ocol];
    float* yb = Ym + ocol;
    #pragma unroll
    for (int r = 0; r < 8; ++r) {
        const int grow = mt * 16 + r + 8 * hh;
        yb[(size_t)grow * DIM] = c[r] + bias;
    }
}

// ---------------------------------------------------------------------------
// Kernel 2: pairwise GAT scores + softmax + attn @ x_val (WMMA).
// One block per (b, h, i-tile of 16 rows).  Stages the full per-(b,h)
// x_aff2 slice ([512][32]) and a TRANSPOSED x_val slice ([32][514]) into the
// 320KB WGP LDS.  leaky_relu(z,0.2) = 0.6*z + 0.4*|z|.
// ---------------------------------------------------------------------------
__global__ __launch_bounds__(256) void gat_attn_kernel(
    const float* __restrict__ Pfc,    // x_aff  stored [BS,N,D] (f = h*DD+d)
    const float* __restrict__ Pfc2,   // x_aff2
    const float* __restrict__ Pval,   // x_val
    const float* __restrict__ adj,    // [BS,N,N]
    const float* __restrict__ aff,    // [H,DD]
    const float* __restrict__ affb,   // [H]
    float* __restrict__ Oattn)        // [BS,N,D]
{
    extern __shared__ float smem[];
    float* s_xa2 = smem;                   // [512][32]   x_aff2 slice   16384
    float* s_xvT = s_xa2 + NN * DD;        // [32][NNP]   x_val^T slice  16448
    float* s_sc  = s_xvT + DD * NNP;       // [16][NNP]   scores -> attn  8224
    float* s_xa  = s_sc  + 16 * NNP;       // [16][32]    x_aff rows       512
    float* s_aff = s_xa  + 16 * DD;        // [2][32]     0.6*aff,0.4*aff   64
    float* s_red = s_aff + 2 * DD;         // [8][256]    WMMA partials   2048

    const int tid  = threadIdx.x;
    const int lane = tid & 31;
    const int wave = tid >> 5;

    const int blk = blockIdx.x;            // BS*H*32 = 1024 blocks
    const int b   = blk >> 9;
    const int h   = (blk >> 5) & 15;
    const int it  = blk & 31;

    // ---- stage x_aff2 (row-major) and x_val (transposed) into LDS ----------
    const float* base2 = Pfc2 + (size_t)b * NN * DIM + h * DD;
    const float* basev = Pval + (size_t)b * NN * DIM + h * DD;
    for (int idx = tid; idx < NN * (DD / 4); idx += 256) {
        const int j  = idx >> 3;
        const int dq = idx & 7;
        ((float4*)s_xa2)[j * 8 + dq] =
            *(const float4*)(base2 + (size_t)j * DIM + dq * 4);
    }
    for (int idx = tid; idx < NN * (DD / 4); idx += 256) {
        const int j  = idx & (NN - 1);     // consecutive tids -> consecutive j
        const int dq = idx >> 9;
        const float4 v = *(const float4*)(basev + (size_t)j * DIM + dq * 4);
        s_xvT[(dq * 4 + 0) * NNP + j] = v.x;
        s_xvT[(dq * 4 + 1) * NNP + j] = v.y;
        s_xvT[(dq * 4 + 2) * NNP + j] = v.z;
        s_xvT[(dq * 4 + 3) * NNP + j] = v.w;
    }
    const float* basea = Pfc + ((size_t)b * NN + it * 16) * DIM + h * DD;
    for (int idx = tid; idx < 16 * DD; idx += 256) {
        const int i = idx >> 5, d = idx & 31;
        s_xa[idx] = basea[(size_t)i * DIM + d];
    }
    if (tid < DD) {
        const float av = aff[h * DD + tid];
        s_aff[tid]      = 0.6f * av;
        s_aff[DD + tid] = 0.4f * av;
    }
    __syncthreads();

    // ---- pairwise additive scores (non-decomposable |.| term) --------------
    // xr and scaled-aff hoisted to registers; yr read as 8x ds_load_b128.
    {
        const float ab = affb[h];
        const int i  = tid & 15;
        const int j0 = tid >> 4;
        const int ig = it * 16 + i;
        const float* adjrow = adj + ((size_t)b * NN + ig) * NN;

        float4 xr4[8], a6[8], a4[8];
        #pragma unroll
        for (int q = 0; q < 8; ++q) {
            xr4[q] = ((const float4*)(s_xa + i * DD))[q];
            a6[q]  = ((const float4*)(s_aff))[q];
            a4[q]  = ((const float4*)(s_aff + DD))[q];
        }
        for (int jj = 0; jj < 32; ++jj) {
            const int j = jj * 16 + j0;
            const float4* yr = (const float4*)(s_xa2 + j * DD);
            float acc = 0.f;
            #pragma unroll
            for (int q = 0; q < 8; ++q) {
                const float4 y = yr[q];
                float t;
                t = xr4[q].x + y.x; acc += a6[q].x * t + a4[q].x * fabsf(t);
                t = xr4[q].y + y.y; acc += a6[q].y * t + a4[q].y * fabsf(t);
                t = xr4[q].z + y.z; acc += a6[q].z * t + a4[q].z * fabsf(t);
                t = xr4[q].w + y.w; acc += a6[q].w * t + a4[q].w * fabsf(t);
            }
            float sc = acc + ab;
            if (adjrow[j] < 1e-5f) sc = -1e9f;
            s_sc[i * NNP + j] = sc;
        }
    }
    __syncthreads();

    // ---- softmax over j: 8 waves x 2 rows, wave32 shuffle reductions -------
    for (int rr = 0; rr < 2; ++rr) {
        const int row = wave * 2 + rr;
        float* sr = s_sc + row * NNP;
        float m = -3.4e38f;
        for (int j = lane; j < NN; j += 32) m = fmaxf(m, sr[j]);
        #pragma unroll
        for (int off = 16; off > 0; off >>= 1) m = fmaxf(m, __shfl_xor(m, off, 32));
        float s = 0.f;
        for (int j = lane; j < NN; j += 32) {
            const float e = __expf(sr[j] - m);
            sr[j] = e;
            s += e;
        }
        #pragma unroll
        for (int off = 16; off > 0; off >>= 1) s += __shfl_xor(s, off, 32);
        const float inv = 1.f / s;
        for (int j = lane; j < NN; j += 32) sr[j] *= inv;
    }
    __syncthreads();

    // ---- out = attn @ x_val  (fp32 WMMA, K split across waves) -------------
    {
        const int nt  = wave & 1;          // d-tile (0: d0..15, 1: d16..31)
        const int kc  = wave >> 1;         // K chunk of 128
        const int l16 = lane & 15;
        const int hh  = lane >> 4;
        const int kbeg = kc * 128;
        const float* acol = s_sc + l16 * NNP + kbeg + 2 * hh;          // attn row
        const float* bcol = s_xvT + (nt * 16 + l16) * NNP + kbeg + 2 * hh; // xval^T row
        v8f c = {0.f, 0.f, 0.f, 0.f, 0.f, 0.f, 0.f, 0.f};
        #pragma unroll 8
        for (int k0 = 0; k0 < 128; k0 += 4) {
            v2f a = *(const v2f*)(acol + k0);
            v2f b = *(const v2f*)(bcol + k0);
            c = __builtin_amdgcn_wmma_f32_16x16x4_f32(false, a, false, b,
                                                      (short)0, c, false, false);
        }
        #pragma unroll
        for (int r = 0; r < 8; ++r)
            s_red[wave * 256 + r * 32 + lane] = c[r];
    }
    __syncthreads();

    // ---- reduce 4 K-partials per d-tile, store [b, it*16+i, h*32+d] --------
    float* orow = Oattn + ((size_t)b * NN + it * 16) * DIM + h * DD;
    for (int idx = tid; idx < 16 * DD; idx += 256) {
        const int i2 = idx >> 5, d = idx & 31;
        const int nt2 = d >> 4, dl = d & 15;
        const int l = (i2 >> 3) * 16 + dl;   // C/D layout inverse
        const int r = i2 & 7;
        float v = 0.f;
        #pragma unroll
        for (int q = 0; q < 4; ++q)
            v += s_red[(nt2 + 2 * q) * 256 + r * 32 + l];
        orow[(size_t)i2 * DIM + d] = v;
    }
}

// ---------------------------------------------------------------------------
// Kernel 3: h = LN(x_aff_flat + attn_out) * g + b, then ReLU.  One block/row.
// ---------------------------------------------------------------------------
__global__ __launch_bounds__(256) void ln_relu_kernel(
    const float* __restrict__ Pfc, const float* __restrict__ Oattn,
    const float* __restrict__ g,   const float* __restrict__ be,
    float* __restrict__ out)
{
    __shared__ float rbuf[8];
    __shared__ float sh_mu, sh_rstd;
    const int row = blockIdx.x;            // 0..1023 (= b*N + n)
    const int tid = threadIdx.x;
    const float* a = Pfc   + (size_t)row * DIM;
    const float* o = Oattn + (size_t)row * DIM;

    const float v0 = a[tid] + o[tid];
    const float v1 = a[tid + 256] + o[tid + 256];

    float s = v0 + v1;
    #pragma unroll
    for (int off = 16; off > 0; off >>= 1) s += __shfl_xor(s, off, 32);
    if ((tid & 31) == 0) rbuf[tid >> 5] = s;
    __syncthreads();
    if (tid == 0) {
        float t = 0.f;
        #pragma unroll
        for (int q = 0; q < 8; ++q) t += rbuf[q];
        sh_mu = t * (1.f / DIM);
    }
    __syncthreads();
    const float mu = sh_mu;
    const float d0 = v0 - mu, d1 = v1 - mu;

    float vs = d0 * d0 + d1 * d1;
    #pragma unroll
    for (int off = 16; off > 0; off >>= 1) vs += __shfl_xor(vs, off, 32);
    if ((tid & 31) == 0) rbuf[tid >> 5] = vs;
    __syncthreads();
    if (tid == 0) {
        float t = 0.f;
        #pragma unroll
        for (int q = 0; q < 8; ++q) t += rbuf[q];
        sh_rstd = rsqrtf(t * (1.f / DIM) + EPSV);
    }
    __syncthreads();
    const float rs = sh_rstd;
    out[(size_t)row * DIM + tid]       = fmaxf(d0 * rs * g[tid]       + be[tid],       0.f);
    out[(size_t)row * DIM + tid + 256] = fmaxf(d1 * rs * g[tid + 256] + be[tid + 256], 0.f);
}

// ---------------------------------------------------------------------------
extern "C" void kernel_launch(void* const* d_in, const int* in_sizes, int n_in,
                              void* d_out, int out_size, void* d_ws, size_t ws_size,
                              hipStream_t stream)
{
    const float* x     = (const float*)d_in[0];
    const float* adj   = (const float*)d_in[1];
    const float* W_fc  = (const float*)d_in[2];
    const float* b_fc  = (const float*)d_in[3];
    const float* W_fc2 = (const float*)d_in[4];
    const float* b_fc2 = (const float*)d_in[5];
    const float* W_val = (const float*)d_in[6];
    const float* b_val = (const float*)d_in[7];
    const float* aff   = (const float*)d_in[8];
    const float* affb  = (const float*)d_in[9];
    const float* ln_g  = (const float*)d_in[10];
    const float* ln_b  = (const float*)d_in[11];
    float* out = (float*)d_out;

    float* ws    = (float*)d_ws;
    const size_t SL = (size_t)BSZ * NN * DIM;     // 524288 floats per slice
    float* Pfc   = ws;
    float* Pfc2  = ws + SL;
    float* Pval  = ws + 2 * SL;
    float* Oattn = ws + 3 * SL;

    proj_gemm_kernel<<<768, 256, 0, stream>>>(x, W_fc, b_fc, W_fc2, b_fc2,
                                              W_val, b_val, Pfc, Pfc2, Pval);

    const size_t smem2 = (size_t)(NN * DD + DD * NNP + 16 * NNP + 16 * DD
                                  + 2 * DD + 8 * 256) * sizeof(float); // ~171 KB
    static bool attr_set = false;
    if (!attr_set) {   // idempotent, host-side only; safe under graph capture
        (void)hipFuncSetAttribute((const void*)gat_attn_kernel,
                                  hipFuncAttributeMaxDynamicSharedMemorySize,
                                  (int)smem2);
        attr_set = true;
    }
    gat_attn_kernel<<<1024, 256, smem2, stream>>>(Pfc, Pfc2, Pval, adj, aff,
                                                  affb, Oattn);

    ln_relu_kernel<<<1024, 256, 0, stream>>>(Pfc, Oattn, ln_g, ln_b, out);
}